// MoE_35278861369681
// MI455X (gfx1250) — compile-verified
//
#include <hip/hip_runtime.h>
#include <hip/hip_bf16.h>
#include <math.h>

// ---------------- problem constants (match reference) ----------------
#define DIMD   128
#define NEXP   64
#define TOPK   2
#define SEQ    1024
#define HID    512
#define NASSIGN (SEQ * TOPK)
#define MAX_TILES (NASSIGN / 16 + NEXP)   // sum ceil(count_e/16) <= 128+64

typedef __attribute__((ext_vector_type(2))) float v2f;
typedef __attribute__((ext_vector_type(4))) float v4f;
typedef __attribute__((ext_vector_type(8))) float v8f;
typedef __attribute__((ext_vector_type(4))) unsigned int u32x4;
typedef __attribute__((ext_vector_type(4))) int i32x4;
typedef __attribute__((ext_vector_type(8))) int i32x8;

#if defined(__has_builtin)
#if __has_builtin(__builtin_amdgcn_tensor_load_to_lds) && \
    __has_builtin(__builtin_amdgcn_s_wait_tensorcnt)
#define USE_TDM 1
#endif
#endif
#ifndef USE_TDM
#define USE_TDM 0
#endif

__device__ __forceinline__ float gelu_exact(float v) {
    // torch F.gelu default: 0.5*v*(1+erf(v/sqrt(2)))
    return 0.5f * v * (1.0f + erff(v * 0.70710678118654752440f));
}

__device__ __forceinline__ v8f wmma_f32(v2f a, v2f b, v8f c) {
    // 8 args: (neg_a, A, neg_b, B, c_mod, C, reuse_a, reuse_b)
    return __builtin_amdgcn_wmma_f32_16x16x4_f32(false, a, false, b, (short)0, c,
                                                 false, false);
}

// -------- dense GEMM + bias + activation --------
// Block = 256 threads (8 waves) computes a 128(M) x 16(N) tile of
// C[M,N] = act(A[M,K] @ B[K,N] + bias[N]).
// B is staged coalesced into LDS in K-chunks of 64 and shared by all 8 waves.
#define KC 64
__global__ void __launch_bounds__(256) moe_dense_gemm_act(
    const float* __restrict__ A, const float* __restrict__ Bm,
    const float* __restrict__ bias, float* __restrict__ C,
    int K, int N, int act /*0=gelu,1=sigmoid*/)
{
    __shared__ float Bs[KC][16];

    const int tid  = threadIdx.x;
    const int wave = tid >> 5;
    const int lane = tid & 31;
    const int r16  = lane & 15;
    const int half = lane >> 4;
    const int kb   = half << 1;           // per-lane K base {0,2}

    const int n0   = blockIdx.x << 4;
    const int m0   = (blockIdx.y << 7) + (wave << 4);

    // coalesced B-stage mapping: thread t loads 4 consecutive floats
    const int skr = tid >> 2;             // 0..63  (k within chunk)
    const int sc4 = (tid & 3) << 2;       // 0,4,8,12 (col within tile)

    const float* arow = A + (size_t)(m0 + r16) * K + kb;

    v8f acc = {};
    for (int k0 = 0; k0 < K; k0 += KC) {
        // stage B[k0..k0+63][n0..n0+15] into LDS (16B per thread, coalesced)
        v4f bl = *(const v4f*)(Bm + (size_t)(k0 + skr) * N + (n0 + sc4));
        __syncthreads();
        *(v4f*)&Bs[skr][sc4] = bl;
        __syncthreads();

        #pragma unroll
        for (int kk = 0; kk < KC; kk += 4) {
            v2f a = *(const v2f*)(arow + k0 + kk);
            v2f b;
            b.x = Bs[kk + kb][r16];
            b.y = Bs[kk + kb + 1][r16];
            acc = wmma_f32(a, b, acc);
        }
    }

    const int nc = n0 + r16;
    const float bs = bias[nc];
    #pragma unroll
    for (int r = 0; r < 8; ++r) {
        const int M = r + (half << 3);
        float v = acc[r] + bs;
        v = (act == 0) ? gelu_exact(v) : (1.0f / (1.0f + expf(-v)));
        C[(size_t)(m0 + M) * N + nc] = v;
    }
}

// -------- top-2 routing per token --------
__global__ void moe_topk_kernel(const float* __restrict__ scores,
                                int* __restrict__ idx, float* __restrict__ w)
{
    int t = blockIdx.x * blockDim.x + threadIdx.x;
    if (t >= SEQ) return;
    const float* s = scores + (size_t)t * NEXP;
    float v1 = -3.0e38f, v2 = -3.0e38f;
    int   i1 = 0, i2 = 0;
    for (int e = 0; e < NEXP; ++e) {
        float v = s[e];
        if (v > v1)      { v2 = v1; i2 = i1; v1 = v; i1 = e; }
        else if (v > v2) { v2 = v; i2 = e; }
    }
    float inv = 1.0f / (v1 + v2);
    idx[t * 2 + 0] = i1;  w[t * 2 + 0] = v1 * inv;
    idx[t * 2 + 1] = i2;  w[t * 2 + 1] = v2 * inv;
}

__global__ void moe_zero_counts(int* counts) {
    if (threadIdx.x < NEXP) counts[threadIdx.x] = 0;
}

__global__ void moe_count_kernel(const int* __restrict__ idx, int* counts) {
    int i = blockIdx.x * blockDim.x + threadIdx.x;
    if (i < NASSIGN) atomicAdd(&counts[idx[i]], 1);
}

__global__ void moe_scan_kernel(const int* __restrict__ counts,
                                int* offsets, int* tileStart, int* cursor)
{
    if (blockIdx.x == 0 && threadIdx.x == 0) {
        int o = 0, t = 0;
        for (int e = 0; e < NEXP; ++e) {
            offsets[e]   = o;
            tileStart[e] = t;
            o += counts[e];
            t += (counts[e] + 15) >> 4;
            cursor[e] = 0;
        }
        offsets[NEXP]   = o;
        tileStart[NEXP] = t;
    }
}

__global__ void moe_fill_kernel(const int* __restrict__ idx, int* cursor,
                                const int* __restrict__ offsets, int* token_list)
{
    int i = blockIdx.x * blockDim.x + threadIdx.x;
    if (i < NASSIGN) {
        int e = idx[i];
        int p = atomicAdd(&cursor[e], 1);
        token_list[offsets[e] + p] = i;    // assignment id = token*2 + k
    }
}

// -------- expert FFN: one workgroup (8 waves) per (expert, 16-token tile) --------
// The 16 gathered token rows of X are brought into LDS with one TDM gather-mode
// tensor_load_to_lds (row stride padded to 130 dwords -> conflict-free A reads).
#define XPITCH 130
__global__ void __launch_bounds__(256) moe_expert_kernel(
    const float* __restrict__ X,
    const float* __restrict__ W1, const float* __restrict__ B1,
    const float* __restrict__ W2, const float* __restrict__ B2,
    const int* __restrict__ token_list, const float* __restrict__ topk_w,
    const int* __restrict__ tileStart, const int* __restrict__ offsets,
    const int* __restrict__ counts, float* __restrict__ y_buf)
{
    __shared__ float Xlds[16][XPITCH];    // gathered tokens, 16 x 128 (+2 pad/row)
    __shared__ float Tlds[16][HID + 4];   // 16 x 516 f32, 8B-aligned rows
    __shared__ int   s_assign[16];
    __shared__ float s_w[16];
    __shared__ int   s_valid[16];

    const int bid = blockIdx.x;
    const int total_tiles = tileStart[NEXP];
    if (bid >= total_tiles) return;       // uniform per block

    int e = 0;
    while (bid >= tileStart[e + 1]) e++;
    const int tile   = bid - tileStart[e];
    const int base   = offsets[e] + (tile << 4);
    const int nvalid = min(16, counts[e] - (tile << 4));

    if (threadIdx.x < 16) {
        int r = threadIdx.x;
        int valid = (r < nvalid) ? 1 : 0;
        int a = token_list[valid ? (base + r) : base];
        s_assign[r] = a;
        s_w[r]      = topk_w[a];
        s_valid[r]  = valid;
    }
    __syncthreads();

#if USE_TDM
    if (threadIdx.x < 32) {               // one TDM issue per workgroup (wave 0)
        // ---- D# group 0: count=1, gather_mode=1 (16-bit idx), lds/global addr, type=2
        unsigned long long ga = (unsigned long long)(uintptr_t)X;
        unsigned ldsa = (unsigned)(uintptr_t)&Xlds[0][0];  // low 32 bits = LDS offset
        u32x4 g0;
        g0[0] = 1u | (1u << 31);
        g0[1] = ldsa;
        g0[2] = (unsigned)(ga & 0xffffffffull);
        g0[3] = (unsigned)((ga >> 32) & 0x01ffffffull) | (2u << 30);
        // ---- D# group 1: data_size=4B, pad 2 dwords every 128 dwords,
        //      tensor_dim0=128, tensor_dim1=1024, tile_dim0=128, tile_dim1=16,
        //      tensor_dim0_stride=128
        i32x8 g1;
        g1[0] = (2 << 16) | (1 << 20) | (6 << 22) | (1 << 25);
        g1[1] = (DIMD & 0xffff) << 16;
        g1[2] = ((SEQ & 0xffff) << 16);
        g1[3] = (DIMD << 16);
        g1[4] = 16;
        g1[5] = DIMD;
        g1[6] = 0;
        g1[7] = 0;
        // ---- D# groups 2/3: 16 row indices (token ids), 16-bit packed
        int tk[16];
        #pragma unroll
        for (int r = 0; r < 16; ++r)
            tk[r] = __builtin_amdgcn_readfirstlane(s_assign[r] >> 1);
        i32x4 g2, g3;
        g2[0] = (tk[0]  & 0xffff) | (tk[1]  << 16);
        g2[1] = (tk[2]  & 0xffff) | (tk[3]  << 16);
        g2[2] = (tk[4]  & 0xffff) | (tk[5]  << 16);
        g2[3] = (tk[6]  & 0xffff) | (tk[7]  << 16);
        g3[0] = (tk[8]  & 0xffff) | (tk[9]  << 16);
        g3[1] = (tk[10] & 0xffff) | (tk[11] << 16);
        g3[2] = (tk[12] & 0xffff) | (tk[13] << 16);
        g3[3] = (tk[14] & 0xffff) | (tk[15] << 16);
        i32x8 gpad = {};                  // extra descriptor slot (6-arg toolchain), zero
        __builtin_amdgcn_tensor_load_to_lds(g0, g1, g2, g3, gpad, 0);
        __builtin_amdgcn_s_wait_tensorcnt(0);
    }
    __syncthreads();
#else
    // fallback: cooperative gather of 16 token rows into LDS
    for (int i = threadIdx.x; i < 16 * DIMD; i += 256) {
        int r = i >> 7, c = i & 127;
        Xlds[r][c] = X[(size_t)(s_assign[r] >> 1) * DIMD + c];
    }
    __syncthreads();
#endif

    const int wave = threadIdx.x >> 5;
    const int lane = threadIdx.x & 31;
    const int r16  = lane & 15;
    const int half = lane >> 4;
    const int kb   = half << 1;

    // ---- Stage A: T[16,512] = gelu( Xg[16,128] @ W1[e]^T + B1[e] ) ----
    const float* xrow = &Xlds[r16][kb];
    #pragma unroll
    for (int nt = 0; nt < 4; ++nt) {
        const int fcol = (((wave << 2) + nt) << 4) + r16;   // 0..511
        const float* wrow = W1 + ((size_t)e * HID + fcol) * DIMD + kb;
        __builtin_prefetch(wrow + DIMD * 16, 0, 1);         // next f-column set
        v8f acc = {};
        #pragma unroll
        for (int k0 = 0; k0 < DIMD; k0 += 4) {
            v2f a = *(const v2f*)(xrow + k0);
            v2f b = *(const v2f*)(wrow + k0);
            acc = wmma_f32(a, b, acc);
        }
        const float bs = B1[(size_t)e * HID + fcol];
        #pragma unroll
        for (int r = 0; r < 8; ++r) {
            const int M = r + (half << 3);
            Tlds[M][fcol] = gelu_exact(acc[r] + bs);
        }
    }
    __syncthreads();

    // ---- Stage B: Y[16,128] = gelu( T @ W2[e]^T + B2[e] ) * gate_w ----
    {
        const int dcol = (wave << 4) + r16;                 // 0..127
        const float* wrow = W2 + ((size_t)e * DIMD + dcol) * HID + kb;
        const float* trow = &Tlds[r16][kb];
        v8f acc = {};
        #pragma unroll
        for (int k0 = 0; k0 < HID; k0 += 4) {
            v2f a = *(const v2f*)(trow + k0);
            v2f b = *(const v2f*)(wrow + k0);
            acc = wmma_f32(a, b, acc);
        }
        const float bs = B2[(size_t)e * DIMD + dcol];
        #pragma unroll
        for (int r = 0; r < 8; ++r) {
            const int M = r + (half << 3);
            if (s_valid[M]) {
                float y = gelu_exact(acc[r] + bs) * s_w[M];
                y_buf[(size_t)s_assign[M] * DIMD + dcol] = y;
            }
        }
    }
}

// -------- final combine: out[t,:] = y(t,k=0) + y(t,k=1) --------
__global__ void moe_combine_kernel(const float* __restrict__ y_buf,
                                   float* __restrict__ out)
{
    int i = blockIdx.x * blockDim.x + threadIdx.x;
    if (i < SEQ * DIMD) {
        int t = i >> 7, d = i & 127;
        out[i] = y_buf[(size_t)(t * 2) * DIMD + d] +
                 y_buf[(size_t)(t * 2 + 1) * DIMD + d];
    }
}

// ---------------------------------------------------------------------
static inline char* align256(char* p) {
    return (char*)(((uintptr_t)p + 255u) & ~(uintptr_t)255u);
}

extern "C" void kernel_launch(void* const* d_in, const int* in_sizes, int n_in,
                              void* d_out, int out_size, void* d_ws, size_t ws_size,
                              hipStream_t stream)
{
    const float* x   = (const float*)d_in[0];
    const float* gw1 = (const float*)d_in[1];
    const float* gb1 = (const float*)d_in[2];
    const float* gw2 = (const float*)d_in[3];
    const float* gb2 = (const float*)d_in[4];
    const float* gw3 = (const float*)d_in[5];
    const float* gb3 = (const float*)d_in[6];
    const float* W1  = (const float*)d_in[7];
    const float* B1  = (const float*)d_in[8];
    const float* W2  = (const float*)d_in[9];
    const float* B2  = (const float*)d_in[10];
    float* out = (float*)d_out;

    // workspace carve-up
    char* p = (char*)d_ws;
    float* h1 = (float*)p;                p = align256(p + sizeof(float) * SEQ * HID);
    float* h2 = (float*)p;                p = align256(p + sizeof(float) * SEQ * HID);
    float* scores = (float*)p;            p = align256(p + sizeof(float) * SEQ * NEXP);
    int*   topk_idx = (int*)p;            p = align256(p + sizeof(int) * NASSIGN);
    float* topk_wt  = (float*)p;          p = align256(p + sizeof(float) * NASSIGN);
    int*   counts   = (int*)p;            p = align256(p + sizeof(int) * NEXP);
    int*   offsets  = (int*)p;            p = align256(p + sizeof(int) * (NEXP + 1));
    int*   tileStart= (int*)p;            p = align256(p + sizeof(int) * (NEXP + 1));
    int*   cursor   = (int*)p;            p = align256(p + sizeof(int) * NEXP);
    int*   token_list = (int*)p;          p = align256(p + sizeof(int) * NASSIGN);
    float* y_buf    = (float*)p;          p = align256(p + sizeof(float) * NASSIGN * DIMD);

    // gate MLP (WMMA f32, LDS-shared B tiles)
    moe_dense_gemm_act<<<dim3(HID / 16, SEQ / 128), 256, 0, stream>>>(
        x, gw1, gb1, h1, DIMD, HID, 0);
    moe_dense_gemm_act<<<dim3(HID / 16, SEQ / 128), 256, 0, stream>>>(
        h1, gw2, gb2, h2, HID, HID, 0);
    moe_dense_gemm_act<<<dim3(NEXP / 16, SEQ / 128), 256, 0, stream>>>(
        h2, gw3, gb3, scores, HID, NEXP, 1);

    // routing
    moe_topk_kernel<<<(SEQ + 255) / 256, 256, 0, stream>>>(scores, topk_idx, topk_wt);
    moe_zero_counts<<<1, 64, 0, stream>>>(counts);
    moe_count_kernel<<<(NASSIGN + 255) / 256, 256, 0, stream>>>(topk_idx, counts);
    moe_scan_kernel<<<1, 1, 0, stream>>>(counts, offsets, tileStart, cursor);
    moe_fill_kernel<<<(NASSIGN + 255) / 256, 256, 0, stream>>>(topk_idx, cursor,
                                                               offsets, token_list);

    // grouped expert FFN (TDM gather + WMMA f32) + combine
    moe_expert_kernel<<<MAX_TILES, 256, 0, stream>>>(
        x, W1, B1, W2, B2, token_list, topk_wt, tileStart, offsets, counts, y_buf);
    moe_combine_kernel<<<(SEQ * DIMD + 255) / 256, 256, 0, stream>>>(y_buf, out);
}